// RejectionSampler_33131377721929
// MI455X (gfx1250) — compile-verified
//
#include <hip/hip_runtime.h>
#include <hip/hip_bf16.h>
#include <stdint.h>

// ---------------------------------------------------------------------------
// Rejection sampler for MI455X (gfx1250).
// Memory-bound: streams ~393 MB (target_probs/draft_probs/exp_uniform) once.
// Floor @ 23.3 TB/s ~= 17 us. No GEMM structure -> WMMA inapplicable; the
// CDNA5-specific path is GLOBAL_LOAD_ASYNC_TO_LDS_B128 (ASYNCcnt) with a
// 4-deep prefetch pipeline: 8192 waves x 4.5 KB in flight ~= 36 MB, covering
// BW*latency (~16 MB) with margin. NT temporal hint: single-pass stream.
// ---------------------------------------------------------------------------

#define INVALID_TOKEN (-1)

constexpr int BB  = 128;
constexpr int KK  = 8;
constexpr int VV  = 32000;
constexpr int TPB = 256;                       // 8 waves (wave32)
constexpr int V4  = VV / 4;                    // 8000 float4 per row
constexpr int FULL_STAGES = V4 / TPB;          // 31 full stages
constexpr int TAIL_V4     = V4 - FULL_STAGES * TPB;  // 64 tail vec4s
constexpr int DEPTH = 4;                       // async pipeline depth

// ---- gfx1250 async global->LDS copy + split wait counters ------------------
#if __has_builtin(__builtin_amdgcn_global_load_async_to_lds_b128)
#define HAVE_ASYNC_BUILTIN 1
#endif

typedef int v4i __attribute__((ext_vector_type(4)));
typedef __attribute__((address_space(1))) v4i* gv4i_p;   // global
typedef __attribute__((address_space(3))) v4i* lv4i_p;   // LDS

__device__ __forceinline__ void compiler_fence() {
  asm volatile("" ::: "memory");
}

__device__ __forceinline__ void async_copy_b128(void* lds_dst, const void* gsrc) {
#if defined(HAVE_ASYNC_BUILTIN)
  __builtin_amdgcn_global_load_async_to_lds_b128(
      (gv4i_p)(uintptr_t)gsrc,
      (lv4i_p)(unsigned)(uintptr_t)lds_dst,
      0, /*cpol: TH_LOAD_NT*/ 1);
#else
  unsigned lds_off = (unsigned)(uintptr_t)lds_dst;
  asm volatile("global_load_async_to_lds_b128 %0, %1, off th:TH_LOAD_NT"
               :: "v"(lds_off), "v"(gsrc)
               : "memory");
#endif
}

#if __has_builtin(__builtin_amdgcn_s_wait_asynccnt)
#define WAIT_ASYNC(n) do { __builtin_amdgcn_s_wait_asynccnt(n); compiler_fence(); } while (0)
#else
#define WAIT_ASYNC(n) asm volatile("s_wait_asynccnt %0" :: "i"(n) : "memory")
#endif

#if __has_builtin(__builtin_amdgcn_s_wait_dscnt)
#define WAIT_DS0() do { __builtin_amdgcn_s_wait_dscnt(0); compiler_fence(); } while (0)
#else
#define WAIT_DS0() asm volatile("s_wait_dscnt 0x0" ::: "memory")
#endif

// ---------------------------------------------------------------------------
// Kernel 1: per-(b,k) row argmax of max(t-d, 1e-5) / (-log(u)) over V.
// (The reference's row normalization is a positive constant -> argmax-invariant.)
// ---------------------------------------------------------------------------

__device__ __forceinline__ void upd(float v, int i, float& bv, int& bi) {
  if (v > bv) { bv = v; bi = i; }     // strict '>' keeps earliest index
}

__device__ __forceinline__ void process4(float4 t, float4 d, float4 u,
                                         int base, float& bv, int& bi) {
  float v0 = __fdividef(fmaxf(t.x - d.x, 1e-5f), -__logf(u.x));
  float v1 = __fdividef(fmaxf(t.y - d.y, 1e-5f), -__logf(u.y));
  float v2 = __fdividef(fmaxf(t.z - d.z, 1e-5f), -__logf(u.z));
  float v3 = __fdividef(fmaxf(t.w - d.w, 1e-5f), -__logf(u.w));
  upd(v0, base + 0, bv, bi);
  upd(v1, base + 1, bv, bi);
  upd(v2, base + 2, bv, bi);
  upd(v3, base + 3, bv, bi);
}

__global__ __launch_bounds__(TPB)
void recovered_argmax_kernel(const float* __restrict__ target_probs, // [B,K+1,V]
                             const float* __restrict__ draft_probs,  // [B,K,V]
                             const float* __restrict__ exp_uniform,  // [B,K,V]
                             int* __restrict__ recovered)            // [B,K]
{
  __shared__ float4 bufT[DEPTH][TPB];
  __shared__ float4 bufD[DEPTH][TPB];
  __shared__ float4 bufU[DEPTH][TPB];
  __shared__ float  red_val[TPB];
  __shared__ int    red_idx[TPB];

  const int row = blockIdx.x;            // 0 .. B*K-1
  const int b   = row / KK;
  const int k   = row - b * KK;
  const int tid = threadIdx.x;

  const float4* rowT = (const float4*)(target_probs + ((size_t)(b * (KK + 1) + k)) * VV);
  const float4* rowD = (const float4*)(draft_probs  + ((size_t)row) * VV);
  const float4* rowU = (const float4*)(exp_uniform  + ((size_t)row) * VV);

  float bestV = -1.0f;
  int   bestI = 0;

  // Each wave's 32 lanes async-copy their own 16B each; each thread later
  // reads back exactly its own bytes -> only per-wave ASYNCcnt waits needed,
  // no workgroup barriers in the streaming loop.
  auto issue = [&](int stage) {
    const int p = stage & (DEPTH - 1);
    const int g = stage * TPB + tid;
    async_copy_b128(&bufT[p][tid], rowT + g);
    async_copy_b128(&bufD[p][tid], rowD + g);
    async_copy_b128(&bufU[p][tid], rowU + g);
  };
  auto consume = [&](int stage) {
    const int p = stage & (DEPTH - 1);
    float4 t = bufT[p][tid];
    float4 d = bufD[p][tid];
    float4 u = bufU[p][tid];
    process4(t, d, u, (stage * TPB + tid) * 4, bestV, bestI);
  };

  // Prologue: fill DEPTH-1 stages.
  issue(0);
  issue(1);
  issue(2);

  // Steady state: stay 3 stages ahead; 9 async ops outstanding beyond the
  // stage being consumed.
  for (int s = 0; s <= FULL_STAGES - DEPTH; ++s) {
    WAIT_DS0();                 // prior reads of the buffer we reuse retired
    issue(s + DEPTH - 1);
    WAIT_ASYNC(3 * (DEPTH - 1));
    consume(s);
  }
  // Drain: last DEPTH-1 stages with decreasing outstanding counts.
  WAIT_ASYNC(6); consume(FULL_STAGES - 3);
  WAIT_ASYNC(3); consume(FULL_STAGES - 2);
  WAIT_ASYNC(0); consume(FULL_STAGES - 1);

  // Tail: 64 vec4s via direct global loads (threads 0..63).
  if (tid < TAIL_V4) {
    const int g = FULL_STAGES * TPB + tid;
    float4 t = rowT[g];
    float4 d = rowD[g];
    float4 u = rowU[g];
    process4(t, d, u, g * 4, bestV, bestI);
  }

  // Block reduction; tie -> smaller index (matches jnp.argmax first-hit).
  red_val[tid] = bestV;
  red_idx[tid] = bestI;
  __syncthreads();
  for (int off = TPB / 2; off > 0; off >>= 1) {
    if (tid < off) {
      float ov = red_val[tid + off];
      int   oi = red_idx[tid + off];
      float mv = red_val[tid];
      if (ov > mv || (ov == mv && oi < red_idx[tid])) {
        red_val[tid] = ov;
        red_idx[tid] = oi;
      }
    }
    __syncthreads();
  }
  if (tid == 0) recovered[row] = red_idx[0];
}

// ---------------------------------------------------------------------------
// Kernel 2: accept/reject chain + splice recovered/bonus token. One thread
// per batch row; a handful of gathers, negligible cost.
// ---------------------------------------------------------------------------
__global__ __launch_bounds__(BB)
void finalize_kernel(const int*   __restrict__ draft_token_ids,  // [B,K]
                     const float* __restrict__ draft_probs,      // [B,K,V]
                     const int*   __restrict__ target_token_ids, // [B,K+1]
                     const float* __restrict__ target_probs,     // [B,K+1,V]
                     const float* __restrict__ uniform,          // [B,K]
                     const int*   __restrict__ recovered,        // [B,K]
                     int*         __restrict__ out)              // [B,K+1]
{
  const int b = blockIdx.x * blockDim.x + threadIdx.x;
  if (b >= BB) return;

  int  outrow[KK + 1];
  int  n_acc = 0;
  bool rejected = false;

  for (int k = 0; k < KK; ++k) {
    const int   id = draft_token_ids[b * KK + k];
    const float pd = draft_probs[((size_t)(b * KK + k)) * VV + id];
    const float pt = target_probs[((size_t)(b * (KK + 1) + k)) * VV + id];
    const float u  = uniform[b * KK + k];
    const bool acc = (u <= pt / pd);
    if (!rejected && acc) {
      outrow[k] = id;
      ++n_acc;
    } else {
      rejected = true;
      outrow[k] = INVALID_TOKEN;
    }
  }
  outrow[KK] = INVALID_TOKEN;

  const int bonus = (n_acc < KK) ? recovered[b * KK + n_acc]
                                 : target_token_ids[b * (KK + 1) + KK];
  outrow[n_acc] = bonus;

  for (int j = 0; j <= KK; ++j) out[b * (KK + 1) + j] = outrow[j];
}

// ---------------------------------------------------------------------------

extern "C" void kernel_launch(void* const* d_in, const int* in_sizes, int n_in,
                              void* d_out, int out_size, void* d_ws, size_t ws_size,
                              hipStream_t stream) {
  const int*   draft_token_ids  = (const int*)d_in[0];   // [B,K]
  const float* draft_probs      = (const float*)d_in[1]; // [B,K,V]
  const int*   target_token_ids = (const int*)d_in[2];   // [B,K+1]
  const float* target_probs     = (const float*)d_in[3]; // [B,K+1,V]
  const float* uniform_samples  = (const float*)d_in[4]; // [B,K]
  const float* exp_uniform      = (const float*)d_in[5]; // [B,K,V]

  int* out       = (int*)d_out;   // [B,K+1] int32
  int* recovered = (int*)d_ws;    // [B,K] scratch

  recovered_argmax_kernel<<<dim3(BB * KK), dim3(TPB), 0, stream>>>(
      target_probs, draft_probs, exp_uniform, recovered);

  finalize_kernel<<<dim3(1), dim3(BB), 0, stream>>>(
      draft_token_ids, draft_probs, target_token_ids, target_probs,
      uniform_samples, recovered, out);
}